// GATModel_70162585748135
// MI455X (gfx1250) — compile-verified
//
#include <hip/hip_runtime.h>
#include <math.h>

#define HEADS     4
#define OUT_FEATS 64
#define HF        256   // HEADS*OUT_FEATS (== IN_FEATS here)
#define NEG_SLOPE 0.2f

typedef float v2f __attribute__((ext_vector_type(2)));
typedef float v8f __attribute__((ext_vector_type(8)));

// ---------------------------------------------------------------------------
// feat[N,256] = A[N,256] @ W[256,256]   (fp32 WMMA, one wave per 16x16 tile)
// A-frag (16x4 f32): lanes 0-15 -> K={0,1}, lanes 16-31 -> K={2,3}
// B-frag (4x16 f32): lanes 0-15 -> K rows 0,1 ; lanes 16-31 -> K rows 2,3
// C/D   (16x16 f32): VGPR g -> M=g (lanes 0-15), M=g+8 (lanes 16-31)
// ---------------------------------------------------------------------------
__global__ __launch_bounds__(256) void gemm_wmma(const float* __restrict__ A,
                                                 const float* __restrict__ W,
                                                 float* __restrict__ out,
                                                 int K, int ntiles)
{
    const int NT_N = HF / 16;                       // 16 column tiles
    int wave = threadIdx.x >> 5;
    int tile = blockIdx.x * 8 + wave;               // wave-uniform
    if (tile >= ntiles) return;                     // uniform per-wave branch
    int tm = tile / NT_N, tn = tile % NT_N;
    int m0 = tm * 16, n0 = tn * 16;
    int lane = threadIdx.x & 31;
    int lm   = lane & 15;
    int hi   = lane >> 4;                           // 0 or 1

    const float* arow = A + (size_t)(m0 + lm) * K + 2 * hi;
    const float* bcol = W + (size_t)(2 * hi) * HF + n0 + lm;

    v8f c = {};
#pragma unroll 4
    for (int k = 0; k < K; k += 4) {
        v2f a, b;
        a.x = arow[k];
        a.y = arow[k + 1];
        b.x = bcol[(size_t)k * HF];
        b.y = bcol[(size_t)k * HF + HF];
        // D = A*B + C : (neg_a, A, neg_b, B, c_mod, C, reuse_a, reuse_b)
        c = __builtin_amdgcn_wmma_f32_16x16x4_f32(false, a, false, b,
                                                  (short)0, c, false, false);
    }

    float* orow = out + (size_t)(m0 + 8 * hi) * HF + n0 + lm;
#pragma unroll
    for (int g = 0; g < 8; ++g)
        orow[(size_t)g * HF] = c[g];
}

// el[n,h] = sum_f feat[n,h,f]*al[h,f];  er likewise (single pass over feat)
__global__ void proj_lr(const float* __restrict__ feat,
                        const float* __restrict__ al,
                        const float* __restrict__ ar,
                        float* __restrict__ el, float* __restrict__ er,
                        int NH)
{
    int idx = blockIdx.x * blockDim.x + threadIdx.x;
    if (idx >= NH) return;
    int n = idx >> 2, h = idx & 3;
    const float* fr = feat + (size_t)n * HF + h * OUT_FEATS;
    const float* alh = al + h * OUT_FEATS;
    const float* arh = ar + h * OUT_FEATS;
    float sl = 0.f, sr = 0.f;
#pragma unroll 8
    for (int f = 0; f < OUT_FEATS; ++f) {
        float v = fr[f];
        sl += v * alh[f];
        sr += v * arh[f];
    }
    el[idx] = sl;
    er[idx] = sr;
}

// per (edge, head): w = exp(leakyrelu(el[src]+er[dst])); denom[dst] += w
__global__ void edge_w(const float* __restrict__ el, const float* __restrict__ er,
                       const int* __restrict__ src, const int* __restrict__ dst,
                       float* __restrict__ wbuf, float* __restrict__ denom,
                       int EH)
{
    int idx = blockIdx.x * blockDim.x + threadIdx.x;
    if (idx >= EH) return;
    int e = idx >> 2, h = idx & 3;
    int s = src[e], d = dst[e];
    float x = el[s * HEADS + h] + er[d * HEADS + h];
    x = (x > 0.f) ? x : NEG_SLOPE * x;
    float w = expf(x);
    wbuf[idx] = w;
    atomicAdd(&denom[d * HEADS + h], w);
}

// 64 lanes per edge (4 floats each): acc[dst] += feat[src] * (w/denom[dst])
__global__ __launch_bounds__(256) void edge_agg(const float* __restrict__ feat,
                                                const float* __restrict__ wbuf,
                                                const float* __restrict__ denom,
                                                const int* __restrict__ src,
                                                const int* __restrict__ dst,
                                                float* __restrict__ acc, int E)
{
    int e = blockIdx.x * 4 + (threadIdx.x >> 6);
    if (e >= E) return;
    int t  = threadIdx.x & 63;
    int h  = t >> 4;
    int f0 = (t & 15) * 4;
    int s = src[e], d = dst[e];
    float coeff = wbuf[e * HEADS + h] / denom[d * HEADS + h];
    const float* fp = feat + (size_t)s * HF + h * OUT_FEATS + f0;
    float*       op = acc  + (size_t)d * HF + h * OUT_FEATS + f0;
#pragma unroll
    for (int j = 0; j < 4; ++j)
        atomicAdd(op + j, fp[j] * coeff);
}

// out = elu(acc + b)
__global__ void finalize(const float* __restrict__ acc, const float* __restrict__ b,
                         float* __restrict__ out, int total)
{
    int idx = blockIdx.x * blockDim.x + threadIdx.x;
    if (idx >= total) return;
    float v = acc[idx] + b[idx & (HF - 1)];
    out[idx] = (v > 0.f) ? v : expm1f(v);
}

// ---------------------------------------------------------------------------
static void run_layer(const float* in, const float* W, const float* al,
                      const float* ar, const float* b, float* layer_out,
                      float* feat, float* accb, float* elb, float* erb,
                      float* denomb, float* wbufb,
                      const int* src, const int* dst, int N, int E,
                      hipStream_t stream)
{
    hipMemsetAsync(accb,   0, (size_t)N * HF    * sizeof(float), stream);
    hipMemsetAsync(denomb, 0, (size_t)N * HEADS * sizeof(float), stream);

    int ntiles = (N / 16) * (HF / 16);
    int gblocks = (ntiles + 7) / 8;
    gemm_wmma<<<gblocks, 256, 0, stream>>>(in, W, feat, HF, ntiles);

    int NH = N * HEADS;
    proj_lr<<<(NH + 255) / 256, 256, 0, stream>>>(feat, al, ar, elb, erb, NH);

    int EH = E * HEADS;
    edge_w<<<(EH + 255) / 256, 256, 0, stream>>>(elb, erb, src, dst, wbufb, denomb, EH);

    edge_agg<<<(E + 3) / 4, 256, 0, stream>>>(feat, wbufb, denomb, src, dst, accb, E);

    finalize<<<(N * HF + 255) / 256, 256, 0, stream>>>(accb, b, layer_out, N * HF);
}

extern "C" void kernel_launch(void* const* d_in, const int* in_sizes, int n_in,
                              void* d_out, int out_size, void* d_ws, size_t ws_size,
                              hipStream_t stream)
{
    const float* h   = (const float*)d_in[0];
    const int*   src = (const int*)d_in[1];
    const int*   dst = (const int*)d_in[2];
    const float* W1  = (const float*)d_in[3];
    const float* al1 = (const float*)d_in[4];
    const float* ar1 = (const float*)d_in[5];
    const float* b1  = (const float*)d_in[6];
    const float* W2  = (const float*)d_in[7];
    const float* al2 = (const float*)d_in[8];
    const float* ar2 = (const float*)d_in[9];
    const float* b2  = (const float*)d_in[10];
    float* out = (float*)d_out;

    int E = in_sizes[1];
    int N = in_sizes[0] / HF;   // IN_FEATS == HF == 256

    float* ws    = (float*)d_ws;
    float* feat  = ws;                                  // N*256
    float* accb  = feat  + (size_t)N * HF;              // N*256
    float* h2    = accb  + (size_t)N * HF;              // N*256
    float* elb   = h2    + (size_t)N * HF;              // N*4
    float* erb   = elb   + (size_t)N * HEADS;           // N*4
    float* denomb= erb   + (size_t)N * HEADS;           // N*4
    float* wbufb = denomb+ (size_t)N * HEADS;           // E*4

    run_layer(h,  W1, al1, ar1, b1, h2,  feat, accb, elb, erb, denomb, wbufb,
              src, dst, N, E, stream);
    run_layer(h2, W2, al2, ar2, b2, out, feat, accb, elb, erb, denomb, wbufb,
              src, dst, N, E, stream);
}